// DeformableUpdatingModel_78151224918748
// MI455X (gfx1250) — compile-verified
//
#include <hip/hip_runtime.h>
#include <hip/hip_bf16.h>

// ---------------------------------------------------------------------------
// DeformableUpdatingModel, collapsed form:
//   out[m,o] = W_dc @ (W_emb @ g[m] + b_emb*s[m]) + b_dc
//   g[m,k]   = (1/4096) * sum_p A[m,p] * i_features[m0,k,p]
//   A[m,p]   = scatter of bilinear tap weights from the (4x downsampled) flow
// Constants for this instance:
//   B=4, num_gop=4, t=3  -> M=48 flows, M0=16 feature maps
//   D=C_in=256, h=w=64 (4096 px), p_motions maps are 256x256
// Roofline: dominant cost is one 67 MB stream of i_features + ~6 MB of flow
// taps  (~4 us at 23.3 TB/s); dense GEMM tail (2 x 48x256x256) runs on
// V_WMMA_F32_16X16X4_F32.
// ---------------------------------------------------------------------------

#define HW    4096      // 64*64
#define HS    256       // source motion map side
#define MTOT  48
#define M0TOT 16
#define DCH   256

typedef __attribute__((ext_vector_type(2))) float v2f;
typedef __attribute__((ext_vector_type(8))) float v8f;

// ---------------------------------------------------------------------------
// Kernel 1: flow resize (exact-4x bilinear => fixed 0.5 weights => 2x2 mean)
// fused with the bilinear-tap scatter into the per-m weight map A (LDS, then
// flushed to global). One block per m.
// ---------------------------------------------------------------------------
__global__ void __launch_bounds__(256)
flow_scatter_kernel(const float* __restrict__ p_motions, float* __restrict__ A_map) {
    __shared__ float A_lds[HW];
    const int m   = blockIdx.x;       // 0..47
    const int tid = threadIdx.x;      // 0..255

    for (int i = tid; i < HW; i += 256) A_lds[i] = 0.0f;
    __syncthreads();

    const float* __restrict__ Py = p_motions + (size_t)m * 2 * HS * HS; // channel 0: dy
    const float* __restrict__ Px = Py + HS * HS;                        // channel 1: dx

    for (int i = tid; i < HW; i += 256) {
        const int py = i >> 6;
        const int px = i & 63;
        // resize: ys = 4*py + 1.5 -> y0 = 4*py+1, wy = 0.5 (never clipped)
        const int sy = 4 * py + 1;
        const int sx = 4 * px + 1;
        const int o00 = sy * HS + sx;
        // (avg of 2x2) * scale(0.25) = 0.0625 * sum
        const float fy = 0.0625f * (Py[o00] + Py[o00 + 1] + Py[o00 + HS] + Py[o00 + HS + 1]);
        const float fx = 0.0625f * (Px[o00] + Px[o00 + 1] + Px[o00 + HS] + Px[o00 + HS + 1]);

        const float y   = (float)py + fy;
        const float x   = (float)px + fx;
        const float y0f = floorf(y);
        const float x0f = floorf(x);
        const float wy  = y - y0f;
        const float wx  = x - x0f;
        const int   y0  = (int)y0f;
        const int   x0  = (int)x0f;

        #pragma unroll
        for (int dy = 0; dy < 2; ++dy) {
            #pragma unroll
            for (int dx = 0; dx < 2; ++dx) {
                const int   yi  = y0 + dy;
                const int   xi  = x0 + dx;
                const float wgt = (dy ? wy : 1.0f - wy) * (dx ? wx : 1.0f - wx);
                if (yi >= 0 && yi < 64 && xi >= 0 && xi < 64)
                    atomicAdd(&A_lds[yi * 64 + xi], wgt);   // ds_add_f32
            }
        }
    }
    __syncthreads();

    float* __restrict__ Am = A_map + (size_t)m * HW;
    for (int i = tid; i < HW; i += 256) Am[i] = A_lds[i];
}

// ---------------------------------------------------------------------------
// Kernel 2: g[m,k] = (1/4096) sum_p A[m,p] * F[m0,k,p]  (+ s[m] on k-tile 0)
// Grid: (16 k-tiles of 16 channels, 16 m0).  i_features streamed exactly once
// from HBM; the 3 A rows sharing an m0 staged in 48 KB LDS and reused across
// all 16 channels of the tile.
// ---------------------------------------------------------------------------
__global__ void __launch_bounds__(256)
weighted_reduce_kernel(const float* __restrict__ ifeat, const float* __restrict__ A_map,
                       float* __restrict__ g, float* __restrict__ svec) {
    __shared__ float Arow[3][HW];
    __shared__ float red[4];

    const int m0  = blockIdx.y;   // 0..15
    const int kt  = blockIdx.x;   // 0..15
    const int tid = threadIdx.x;  // 0..255

    const float* __restrict__ Abase = A_map + (size_t)m0 * 3 * HW;
    for (int i = tid; i < 3 * HW; i += 256) (&Arow[0][0])[i] = Abase[i];
    __syncthreads();

    // s[m] (only one k-tile needs to do it)
    if (kt == 0) {
        float s0 = 0.f, s1 = 0.f, s2 = 0.f;
        for (int i = tid; i < HW; i += 256) { s0 += Arow[0][i]; s1 += Arow[1][i]; s2 += Arow[2][i]; }
        if (tid < 3) red[tid] = 0.0f;
        __syncthreads();
        #pragma unroll
        for (int off = 16; off > 0; off >>= 1) {
            s0 += __shfl_down(s0, off, 32);
            s1 += __shfl_down(s1, off, 32);
            s2 += __shfl_down(s2, off, 32);
        }
        if ((tid & 31) == 0) { atomicAdd(&red[0], s0); atomicAdd(&red[1], s1); atomicAdd(&red[2], s2); }
        __syncthreads();
        if (tid < 3) svec[m0 * 3 + tid] = red[tid] * (1.0f / (float)HW);
        __syncthreads();
    }

    for (int c = 0; c < 16; ++c) {
        const int k = kt * 16 + c;
        const float* __restrict__ F = ifeat + ((size_t)m0 * DCH + k) * HW;
        float a0 = 0.f, a1 = 0.f, a2 = 0.f;
        for (int i = tid; i < HW; i += 256) {
            const float f = F[i];
            a0 = fmaf(f, Arow[0][i], a0);
            a1 = fmaf(f, Arow[1][i], a1);
            a2 = fmaf(f, Arow[2][i], a2);
        }
        if (tid < 3) red[tid] = 0.0f;
        __syncthreads();
        #pragma unroll
        for (int off = 16; off > 0; off >>= 1) {
            a0 += __shfl_down(a0, off, 32);
            a1 += __shfl_down(a1, off, 32);
            a2 += __shfl_down(a2, off, 32);
        }
        if ((tid & 31) == 0) { atomicAdd(&red[0], a0); atomicAdd(&red[1], a1); atomicAdd(&red[2], a2); }
        __syncthreads();
        if (tid < 3) g[(size_t)(m0 * 3 + tid) * DCH + k] = red[tid] * (1.0f / (float)HW);
        __syncthreads();
    }
}

// ---------------------------------------------------------------------------
// Kernels 3/4: C(48x256) = A(48x256) * B(256x256) + epilogue via
// V_WMMA_F32_16X16X4_F32.  B[k][n] = W[n*256+k] (W row-major (n,k)).
// One wave per 16x16 output tile; grid (16 n-tiles, 3 m-tiles); 64 chained
// WMMAs along K.  EXEC all-ones (full wave, uniform control flow).
// Fragment layouts per ISA 7.12.2:
//   A 16x4 f32 : lanes 0-15 hold M=lane, VGPR0/1 = K0/K1; lanes 16-31 = K2/K3
//   B 4x16 f32 : VGPR0 = {K0 | K2} rows, VGPR1 = {K1 | K3}, lanes = N
//   C/D        : VGPR j = {M=j (lanes 0-15) | M=j+8 (lanes 16-31)}, lanes = N
// RANK1 is a compile-time flag -> straight-line epilogue, no per-lane branch:
//   RANK1=true  : + bias[n]*svec[m]   (b_emb ⊗ s term)
//   RANK1=false : + bias[n]           (b_dc)
// ---------------------------------------------------------------------------
template <bool RANK1>
__global__ void __launch_bounds__(32)
wmma_gemm48_kernel(const float* __restrict__ Amat,   // 48 x 256 row-major
                   const float* __restrict__ W,      // 256 x 256 row-major (n,k)
                   const float* __restrict__ bias,   // 256
                   const float* __restrict__ svec,   // 48 (used iff RANK1)
                   float* __restrict__ Cmat) {       // 48 x 256 row-major
    const int tileN = blockIdx.x;        // 0..15
    const int tileM = blockIdx.y;        // 0..2
    const int lane  = threadIdx.x;       // 0..31
    const int half  = lane >> 4;         // 0 | 1
    const int l15   = lane & 15;

    const int mA = tileM * 16 + l15;     // A-fragment row (lane = M)
    const int nB = tileN * 16 + l15;     // B-fragment col (lane = N)

    v8f acc = {0.f, 0.f, 0.f, 0.f, 0.f, 0.f, 0.f, 0.f};

    for (int k = 0; k < DCH; k += 4) {
        const int kA = k + half * 2;
        v2f afrag, bfrag;
        afrag.x = Amat[mA * DCH + kA + 0];
        afrag.y = Amat[mA * DCH + kA + 1];
        bfrag.x = W[nB * DCH + kA + 0];      // B[kA+0][nB]
        bfrag.y = W[nB * DCH + kA + 1];      // B[kA+1][nB]
        acc = __builtin_amdgcn_wmma_f32_16x16x4_f32(
            /*neg_a=*/false, afrag, /*neg_b=*/false, bfrag,
            /*c_mod=*/(short)0, acc, /*reuse_a=*/false, /*reuse_b=*/false);
    }

    const int   n    = tileN * 16 + l15;
    const float bn   = bias[n];
    const int   mrow0 = tileM * 16 + half * 8;   // rows mrow0..mrow0+7 (VGPR j)

    if (RANK1) {
        float sv[8];
        #pragma unroll
        for (int j = 0; j < 8; ++j) sv[j] = svec[mrow0 + j];
        #pragma unroll
        for (int j = 0; j < 8; ++j)
            Cmat[(size_t)(mrow0 + j) * DCH + n] = fmaf(bn, sv[j], acc[j]);
    } else {
        #pragma unroll
        for (int j = 0; j < 8; ++j)
            Cmat[(size_t)(mrow0 + j) * DCH + n] = acc[j] + bn;
    }
}

// ---------------------------------------------------------------------------
// Launch
// ---------------------------------------------------------------------------
extern "C" void kernel_launch(void* const* d_in, const int* in_sizes, int n_in,
                              void* d_out, int out_size, void* d_ws, size_t ws_size,
                              hipStream_t stream) {
    // setup_inputs order: imgs, i_features, p_motions, W_emb, b_emb, W_dc, b_dc
    const float* ifeat = (const float*)d_in[1];
    const float* pmot  = (const float*)d_in[2];
    const float* W_emb = (const float*)d_in[3];
    const float* b_emb = (const float*)d_in[4];
    const float* W_dc  = (const float*)d_in[5];
    const float* b_dc  = (const float*)d_in[6];
    float* out = (float*)d_out;                 // (4,4,3,256) == (48,256) row-major

    char* ws = (char*)d_ws;
    float* A_map = (float*)ws;                                                               // 48*4096
    float* g     = (float*)(ws + (size_t)(MTOT * HW) * sizeof(float));                       // 48*256
    float* svec  = (float*)(ws + (size_t)(MTOT * HW + MTOT * DCH) * sizeof(float));          // 48 (+pad)
    float* E     = (float*)(ws + (size_t)(MTOT * HW + MTOT * DCH + 64) * sizeof(float));     // 48*256

    // 1) flow (4x bilinear resize) + bilinear-tap scatter -> A_map
    flow_scatter_kernel<<<MTOT, 256, 0, stream>>>(pmot, A_map);

    // 2) g[m,k] = <A[m], F[m0,k]>/4096 ; s[m] = sum(A[m])/4096
    weighted_reduce_kernel<<<dim3(16, 16), 256, 0, stream>>>(ifeat, A_map, g, svec);

    // 3) E = g @ W_emb^T + b_emb ⊗ s          (WMMA f32 16x16x4, rank-1 epilogue)
    wmma_gemm48_kernel<true><<<dim3(16, 3), 32, 0, stream>>>(g, W_emb, b_emb, svec, E);

    // 4) out = E @ W_dc^T + b_dc              (WMMA f32 16x16x4, bias epilogue)
    wmma_gemm48_kernel<false><<<dim3(16, 3), 32, 0, stream>>>(E, W_dc, b_dc, nullptr, out);
}